// AbstractRelu_76751065579631
// MI455X (gfx1250) — compile-verified
//
#include <hip/hip_runtime.h>

// AbstractRelu (zonotope) for x[E=512, N=65536] f32.
// Phase 1: sum_abs[n] = sum_{e>=1} |x[e,n]|   (async-LDS staged streaming reduce)
// Phase 2: out[e,n] = f(x[e,n]; lb,ub,alpha)  (elementwise, NT load/store)
//
// Roofline: 128 MB read + 128 MB write => ~11 us at 23.3 TB/s. Phase 2's
// re-read of x hits the 192 MB L2 (populated by phase 1) as long as the
// output stream is written non-temporally.

typedef __attribute__((ext_vector_type(4))) float float4v;
typedef __attribute__((ext_vector_type(4))) int   int4v;

#define E_ROWS 512
#define TPB 256
#define ROWS_PER_CHUNK 64
#define DEPTH 8  // async pipeline depth: 8 x 32 lanes x 16B = 4 KB in flight/wave

#if defined(__gfx1250__) &&                                            \
    __has_builtin(__builtin_amdgcn_global_load_async_to_lds_b128) &&   \
    __has_builtin(__builtin_amdgcn_s_wait_asynccnt)
#define USE_ASYNC_LDS 1
#else
#define USE_ASYNC_LDS 0
#endif

#define COMPILER_MEM_FENCE() __asm__ __volatile__("" ::: "memory")

#if USE_ASYNC_LDS
// Builtin signature (probe-confirmed): v4i pointers in addrspace(1)/(3).
typedef int4v __attribute__((address_space(1))) g_int4v;  // global
typedef int4v __attribute__((address_space(3))) l_int4v;  // LDS
#endif

// ---------------------------------------------------------------- phase 1 --
__global__ __launch_bounds__(TPB) void zono_sumabs(const float* __restrict__ x,
                                                   float* __restrict__ sum_abs,
                                                   int N) {
    const int col0 = blockIdx.x * (TPB * 4) + threadIdx.x * 4;
    int rs = blockIdx.y * ROWS_PER_CHUNK;
    const int re = rs + ROWS_PER_CHUNK;  // E_ROWS % ROWS_PER_CHUNK == 0
    if (rs == 0) rs = 1;  // row 0 is the center term, excluded from sum_abs
    // Note: re - rs >= ROWS_PER_CHUNK - 1 >= DEPTH always -> no prologue guards.

    const float* base = x + col0;
    float4v acc = {0.f, 0.f, 0.f, 0.f};

#if USE_ASYNC_LDS
    __shared__ float4v buf[DEPTH][TPB];

    // Prologue: fill the pipeline (each issue bumps ASYNCcnt).
#pragma unroll
    for (int i = 0; i < DEPTH; ++i) {
        __builtin_amdgcn_global_load_async_to_lds_b128(
            (g_int4v*)(base + (size_t)(rs + i) * N),
            (l_int4v*)&buf[i][threadIdx.x], 0, 0);
    }

    int r = rs;
    const int steady_end = re - DEPTH;
    for (; r < steady_end; ++r) {
        // Oldest in-flight copy has landed in LDS (in-order completion).
        __builtin_amdgcn_s_wait_asynccnt(DEPTH - 1);
        COMPILER_MEM_FENCE();
        float4v v = buf[(r - rs) % DEPTH][threadIdx.x];
        acc += __builtin_elementwise_abs(v);
        // Fence: the reissue below must stay after the consumption of v, so
        // the backend's s_wait_dscnt (data dep on v) precedes the async issue
        // -> no WAR race on the LDS slot.
        COMPILER_MEM_FENCE();
        __builtin_amdgcn_global_load_async_to_lds_b128(
            (g_int4v*)(base + (size_t)(r + DEPTH) * N),
            (l_int4v*)&buf[(r - rs) % DEPTH][threadIdx.x], 0, 0);
    }
    __builtin_amdgcn_s_wait_asynccnt(0);  // drain the tail
    COMPILER_MEM_FENCE();
    for (; r < re; ++r) {
        float4v v = buf[(r - rs) % DEPTH][threadIdx.x];
        acc += __builtin_elementwise_abs(v);
    }
#else
    for (int r = rs; r < re; ++r) {
        float4v v = *(const float4v*)(base + (size_t)r * N);
        acc += __builtin_elementwise_abs(v);
    }
#endif

    // 8 row-chunks contend per column: trivial for the L2 atomic units.
    atomicAdd(&sum_abs[col0 + 0], acc.x);
    atomicAdd(&sum_abs[col0 + 1], acc.y);
    atomicAdd(&sum_abs[col0 + 2], acc.z);
    atomicAdd(&sum_abs[col0 + 3], acc.w);
}

// ---------------------------------------------------------------- phase 2 --
__global__ __launch_bounds__(TPB) void zono_apply(const float* __restrict__ x,
                                                  const float* __restrict__ sum_abs,
                                                  float* __restrict__ out,
                                                  int N) {
    const int c = (blockIdx.x * TPB + threadIdx.x) * 4;
    const int e = blockIdx.y;  // row index: uniform per block (no int division)

    // Hot, tiny, L2-resident: keep default RT temporal hints.
    const float4v sa = *(const float4v*)(sum_abs + c);
    const float4v x0 = *(const float4v*)(x + c);
    // Last use of the bulk x stream: non-temporal.
    const float4v v =
        __builtin_nontemporal_load((const float4v*)(x + (size_t)e * N + c));

    float4v res;
#pragma unroll
    for (int k = 0; k < 4; ++k) {
        const float lb = x0[k] - sa[k];
        const float ub = x0[k] + sa[k];
        const bool crossing = (lb <= 0.f) && (ub >= 0.f);
        const float alpha = 1.f - lb;  // faithful to source: alpha = 1 - lb
        float r;
        if (crossing) {
            r = (e == 0) ? (alpha * v[k] - alpha * lb * 0.5f) : (alpha * v[k]);
        } else {
            r = v[k];
        }
        res[k] = (ub <= 0.f) ? 0.f : r;  // dead columns zeroed last
    }
    // NT store: don't let the 128 MB output evict x from the 192 MB L2.
    __builtin_nontemporal_store(res, (float4v*)(out + (size_t)e * N + c));
}

// ----------------------------------------------------------------- launch --
extern "C" void kernel_launch(void* const* d_in, const int* in_sizes, int n_in,
                              void* d_out, int out_size, void* d_ws, size_t ws_size,
                              hipStream_t stream) {
    const float* x = (const float*)d_in[0];
    float* out = (float*)d_out;
    float* sum_abs = (float*)d_ws;  // N floats = 256 KB of scratch

    const int total = in_sizes[0];
    const int N = total / E_ROWS;  // 65536

    (void)hipMemsetAsync(sum_abs, 0, (size_t)N * sizeof(float), stream);

    dim3 gridB(N / (TPB * 4), E_ROWS / ROWS_PER_CHUNK);
    zono_sumabs<<<gridB, TPB, 0, stream>>>(x, sum_abs, N);

    dim3 gridC(N / (TPB * 4), E_ROWS);
    zono_apply<<<gridC, TPB, 0, stream>>>(x, sum_abs, out, N);
}